// HierarchicalRouter_8555574854213
// MI455X (gfx1250) — compile-verified
//
#include <hip/hip_runtime.h>
#include <hip/hip_bf16.h>

typedef __attribute__((ext_vector_type(2))) float v2f;
typedef __attribute__((ext_vector_type(8))) float v8f;

#define H     1024
#define Dh    512
#define NE    8
#define NTOK  8192
#define KB    256
#define NPAN  (H / KB)   // 4 K-panels
#define RS    260   // padded LDS row stride (floats): (260*r + c) banks distinct over r=0..15
#define HS    520   // padded LDS row stride for router h tile

// --- CDNA5 async global->LDS copy (ASYNCcnt-tracked, per-lane addressing) ---
__device__ __forceinline__ void async_load_b128(unsigned lds_off, const void* gptr) {
  asm volatile("global_load_async_to_lds_b128 %0, %1, off"
               :: "v"(lds_off), "v"(gptr)
               : "memory");
}
template <int N>
__device__ __forceinline__ void wait_async() {
  asm volatile("s_wait_asynccnt %0" :: "i"(N) : "memory");
}
__device__ __forceinline__ unsigned lds_byte_off(const void* p) {
  // generic LDS pointer: low 32 bits == workgroup-relative LDS byte offset
  return (unsigned)(uintptr_t)p;
}

// ---------------------------------------------------------------------------
// Router: h = relu(X@w1 + b1) via f32 WMMA, logits = h@w2 + b2, argmax -> idx
// block = 256 threads (8 waves), one 16-token tile; wave w owns cols
// [w*64, w*64+64) = 4 N-tiles of 16.
// ---------------------------------------------------------------------------
__global__ __launch_bounds__(256) void router_kernel(
    const float* __restrict__ x, const float* __restrict__ w1,
    const float* __restrict__ b1, const float* __restrict__ w2,
    const float* __restrict__ b2, int* __restrict__ expert_idx)
{
  __shared__ float ldsX[16 * RS];
  __shared__ float ldsH[16 * HS];
  __shared__ float lgt[16 * NE];
  const int tid  = threadIdx.x;
  const int lane = tid & 31;
  const int wave = tid >> 5;
  const int row0 = blockIdx.x * 16;

  const int arow = lane & 15;         // A fragment: M = lane%16
  const int koff = (lane >> 4) * 2;   // lanes 0-15 -> K=0,1 ; lanes 16-31 -> K=2,3
  const int ncl  = lane & 15;         // B/D fragment: N = lane%16

  v8f acc[4] = {};

  for (int kb = 0; kb < H; kb += KB) {
    __syncthreads();
    { // async-stage X[:, kb:kb+256) into padded LDS (16 rows x 256 f32)
      const int r = tid >> 4, c = tid & 15;
      const float* g = x + (size_t)(row0 + r) * H + kb;
      const unsigned lbase = lds_byte_off(ldsX + r * RS);
      #pragma unroll
      for (int j = 0; j < 4; ++j)
        async_load_b128(lbase + (unsigned)(c + j * 16) * 16u,
                        g + (c + j * 16) * 4);
      wait_async<0>();
    }
    __syncthreads();
    for (int k2 = 0; k2 < KB; k2 += 4) {
      v2f a;
      a.x = ldsX[arow * RS + k2 + koff];
      a.y = ldsX[arow * RS + k2 + koff + 1];
      const float* wrow = w1 + (size_t)(kb + k2 + koff) * Dh;
      #pragma unroll
      for (int t = 0; t < 4; ++t) {
        const int n = wave * 64 + t * 16 + ncl;
        v2f b;
        b.x = wrow[n];        // w1[k+koff,   n]
        b.y = wrow[Dh + n];   // w1[k+koff+1, n]
        acc[t] = __builtin_amdgcn_wmma_f32_16x16x4_f32(
            false, a, false, b, (short)0, acc[t], false, false);
      }
    }
  }
  __syncthreads();
  // relu(h) -> LDS (D layout: VGPR r -> rows r / r+8)
  const int rbase = (lane >> 4) * 8;
  #pragma unroll
  for (int t = 0; t < 4; ++t) {
    const int col  = wave * 64 + t * 16 + ncl;
    const float bi = b1[col];
    #pragma unroll
    for (int r = 0; r < 8; ++r) {
      float v = acc[t][r] + bi;
      ldsH[(rbase + r) * HS + col] = v > 0.f ? v : 0.f;
    }
  }
  __syncthreads();
  // logits: 128 threads, (row, expert) pair each -> 512-dot
  if (tid < 16 * NE) {
    const int r = tid >> 3, e = tid & 7;
    float s = b2[e];
    for (int d = 0; d < Dh; ++d) s += ldsH[r * HS + d] * w2[d * NE + e];
    lgt[r * NE + e] = s;
  }
  __syncthreads();
  if (tid < 16) {
    float best = -3.4e38f;
    int bi = 0;
    #pragma unroll
    for (int e = 0; e < NE; ++e) {
      const float s = lgt[tid * NE + e];
      if (s > best) { best = s; bi = e; }  // '>' keeps first max like jnp.argmax
    }
    expert_idx[row0 + tid] = bi;
  }
}

__global__ void zero_counts(int* counts) { counts[threadIdx.x] = 0; }

__global__ void build_lists(const int* __restrict__ expert_idx,
                            int* __restrict__ counts, int* __restrict__ lists)
{
  const int t = blockIdx.x * blockDim.x + threadIdx.x;
  if (t < NTOK) {
    const int e   = expert_idx[t];
    const int pos = atomicAdd(&counts[e], 1);
    lists[e * NTOK + pos] = t;
  }
}

// ---------------------------------------------------------------------------
// Expert apply: expert e = blockIdx.y, column chunk blockIdx.x (128 cols,
// wave w owns one 16-col N-tile); loop over the expert's 16-token tiles.
// Double-buffered LDS X-panels; async copy of panel p+1 overlaps WMMA on p.
// out = a_le*(Xg @ W_base[l,e] + b_base[l,e]); final += hw[l]*out; xnext = out
// ---------------------------------------------------------------------------
__global__ __launch_bounds__(256) void expert_kernel(
    const float* __restrict__ x, const float* __restrict__ Wb,
    const float* __restrict__ bb, const float* __restrict__ assign,
    const float* __restrict__ hwv, const int* __restrict__ counts,
    const int* __restrict__ lists, float* __restrict__ xnext,
    float* __restrict__ fout, int layer)
{
  __shared__ float ldsX[2][16 * RS];
  __shared__ int   toks[16];
  const int tid  = threadIdx.x;
  const int lane = tid & 31;
  const int wave = tid >> 5;
  const int e    = blockIdx.y;
  const int cnt    = counts[e];
  const int ntiles = (cnt + 15) >> 4;
  const int n0     = blockIdx.x * 128 + wave * 16;
  const float* W   = Wb + (size_t)e * H * H;

  // hw = softmax(hierarchical_weights)[layer]  (L == 3)
  const float h0 = hwv[0], h1 = hwv[1], h2 = hwv[2];
  const float mx = fmaxf(h0, fmaxf(h1, h2));
  const float s0 = __expf(h0 - mx), s1 = __expf(h1 - mx), s2 = __expf(h2 - mx);
  const float hw = (layer == 0 ? s0 : layer == 1 ? s1 : s2) / (s0 + s1 + s2);
  const float a_le = assign[e];

  const int arow  = lane & 15;
  const int koff  = (lane >> 4) * 2;
  const int ncol  = n0 + (lane & 15);
  const int rbase = (lane >> 4) * 8;
  const float bias = bb[e * H + ncol];

  const int gr = tid >> 4, gc = tid & 15;  // staging role: row, chunk-in-row
  // issue one K-panel (16 rows x 256 f32) of gathered token rows -> LDS buffer
  auto issue_panel = [&](int kb, int buf) {
    const int tok = toks[gr];
    const float* g = x + (size_t)(tok < 0 ? 0 : tok) * H + kb;
    const unsigned lbase = lds_byte_off(&ldsX[buf][gr * RS]);
    #pragma unroll
    for (int j = 0; j < 4; ++j)
      async_load_b128(lbase + (unsigned)(gc + j * 16) * 16u,
                      g + (gc + j * 16) * 4);
  };

  for (int t = 0; t < ntiles; ++t) {
    if (tid < 16) {
      const int i = t * 16 + tid;
      toks[tid] = (i < cnt) ? lists[e * NTOK + i] : -1;
    }
    __syncthreads();          // toks visible; prev tile's LDS consumers done
    issue_panel(0, 0);        // prologue: panel 0 in flight
    v8f acc = {};
    #pragma unroll
    for (int p = 0; p < NPAN; ++p) {
      __syncthreads();        // all waves done reading buf[(p+1)&1] (panel p-1)
      if (p + 1 < NPAN) {
        issue_panel((p + 1) * KB, (p + 1) & 1);  // overlap copy with compute
        wait_async<4>();      // 4 newest (panel p+1) may stay in flight;
                              // in-order completion => panel p has landed
      } else {
        wait_async<0>();
      }
      __syncthreads();        // panel p visible to all waves
      if (p + 1 < NPAN)       // hint next K-panel of W (global_prefetch_b8)
        __builtin_prefetch(W + (size_t)(p + 1) * KB * H + ncol, 0, 1);
      const float* ldsP = ldsX[p & 1];
      const int kbase = p * KB;
      for (int k2 = 0; k2 < KB; k2 += 4) {
        v2f a, b;
        a.x = ldsP[arow * RS + k2 + koff];
        a.y = ldsP[arow * RS + k2 + koff + 1];
        const float* wrow = W + (size_t)(kbase + k2 + koff) * H + ncol;
        b.x = wrow[0];
        b.y = wrow[H];
        acc = __builtin_amdgcn_wmma_f32_16x16x4_f32(
            false, a, false, b, (short)0, acc, false, false);
      }
    }
    #pragma unroll
    for (int r = 0; r < 8; ++r) {
      const int tok = toks[rbase + r];
      if (tok >= 0) {
        const float v    = a_le * (acc[r] + bias);
        const size_t off = (size_t)tok * H + ncol;
        xnext[off] = v;
        fout[off]  = (layer == 0) ? hw * v : fout[off] + hw * v;
      }
    }
    __syncthreads();          // epilogue reads of toks done before next tile
  }
}

// ---------------------------------------------------------------------------
extern "C" void kernel_launch(void* const* d_in, const int* in_sizes, int n_in,
                              void* d_out, int out_size, void* d_ws, size_t ws_size,
                              hipStream_t stream) {
  (void)in_sizes; (void)n_in; (void)out_size; (void)ws_size;
  const float* x0   = (const float*)d_in[0];  // (B,S,H)
  const float* w1   = (const float*)d_in[1];  // (L,H,512)
  const float* b1   = (const float*)d_in[2];  // (L,512)
  const float* w2   = (const float*)d_in[3];  // (L,512,8)
  const float* b2   = (const float*)d_in[4];  // (L,8)
  const float* lea  = (const float*)d_in[5];  // (L,8)
  const float* hwv  = (const float*)d_in[6];  // (L,)
  const float* Wb   = (const float*)d_in[7];  // (L,8,H,H)
  const float* bb   = (const float*)d_in[8];  // (L,8,H)
  float* fout = (float*)d_out;

  // workspace: two N*H f32 ping-pong buffers + routing metadata
  float* bufA = (float*)d_ws;
  float* bufB = bufA + (size_t)NTOK * H;
  int* expert_idx = (int*)(bufB + (size_t)NTOK * H);
  int* counts     = expert_idx + NTOK;
  int* lists      = counts + NE;

  const float* xin[3]  = { x0, bufA, bufB };
  float*       xout[3] = { bufA, bufB, bufA };

  for (int l = 0; l < 3; ++l) {
    router_kernel<<<NTOK / 16, 256, 0, stream>>>(
        xin[l], w1 + (size_t)l * H * Dh, b1 + (size_t)l * Dh,
        w2 + (size_t)l * Dh * NE, b2 + (size_t)l * NE, expert_idx);
    zero_counts<<<1, NE, 0, stream>>>(counts);
    build_lists<<<NTOK / 256, 256, 0, stream>>>(expert_idx, counts, lists);
    expert_kernel<<<dim3(8, NE), 256, 0, stream>>>(
        xin[l], Wb + (size_t)l * NE * H * H, bb + (size_t)l * NE * H,
        lea + (size_t)l * NE, hwv, counts, lists, xout[l], fout, l);
  }
}